// TransformerBaseline_53257594470469
// MI455X (gfx1250) — compile-verified
//
#include <hip/hip_runtime.h>

#define NN 100000   // divisible by 16 -> exact WMMA tiling, no edge guards needed
#define EE 1600000

typedef __attribute__((ext_vector_type(2))) float v2f;
typedef __attribute__((ext_vector_type(8))) float v8f;

// ---------------------------------------------------------------- utilities

__global__ void gt_zero_kernel(float* __restrict__ p, int n) {
  int i = blockIdx.x * blockDim.x + threadIdx.x;
  if (i < n) p[i] = 0.0f;
}

// Pack [Wq|Wk|Wv|Wskip] rows 0..31, bias row 32, We rows 33..36 into 37x128.
__global__ void gt_pack_kernel(float* __restrict__ wpack,
                               const float* __restrict__ Wq, const float* __restrict__ bq,
                               const float* __restrict__ Wk, const float* __restrict__ bk,
                               const float* __restrict__ Wv, const float* __restrict__ bv,
                               const float* __restrict__ We,
                               const float* __restrict__ Wskip, const float* __restrict__ bskip,
                               int din, int hc, int skip_out) {
  int idx = blockIdx.x * blockDim.x + threadIdx.x;
  if (idx >= 37 * 128) return;
  int r = idx >> 7, c = idx & 127;
  int h3 = 3 * hc;
  float val = 0.0f;
  if (r < din) {
    if (c < hc)            val = Wq[r * hc + c];
    else if (c < 2 * hc)   val = Wk[r * hc + (c - hc)];
    else if (c < h3)       val = Wv[r * hc + (c - 2 * hc)];
    else if (c < h3 + skip_out) val = Wskip[r * skip_out + (c - h3)];
  } else if (r == 32) {
    if (c < hc)            val = bq[c];
    else if (c < 2 * hc)   val = bk[c - hc];
    else if (c < h3)       val = bv[c - 2 * hc];
    else if (c < h3 + skip_out) val = bskip[c - h3];
  } else if (r >= 33) {
    int t = r - 33;
    if (c < hc) val = We[t * hc + c];
  }
  wpack[idx] = val;
}

// ------------------------------------------------- fused projection (WMMA f32)
// proj[n, 0:128] = [q | k | v | skip] row for node n (bias included).
// 8 waves/block; wave w computes the 16x16 tile at cols [16w, 16w+16).
// Requires Nn % 16 == 0 (true for NN): no guards, EXEC stays all-ones.
template <int DIN>
__global__ void gt_proj_wmma_kernel(const float* __restrict__ h,
                                    const float* __restrict__ wpack,
                                    float* __restrict__ proj) {
  const int lane = threadIdx.x & 31;
  const int wave = threadIdx.x >> 5;     // 0..7
  const int half = lane >> 4;            // 0/1
  const int mn   = lane & 15;
  const int rowbase = blockIdx.x * 16;

  const int node = rowbase + mn;         // A-fragment row
  const int col  = wave * 16 + mn;       // B/D column

  // C init = bias (depends on column only)
  const float bias = wpack[32 * 128 + col];
  v8f acc;
#pragma unroll
  for (int r = 0; r < 8; ++r) acc[r] = bias;

  const float* __restrict__ arow = h + (size_t)node * DIN;
#pragma unroll
  for (int k0 = 0; k0 < DIN; k0 += 4) {
    const int k = k0 + 2 * half;         // lanes 0-15 -> K=k0,k0+1 ; 16-31 -> k0+2,k0+3
    v2f a;  a[0] = arow[k];              a[1] = arow[k + 1];
    v2f b;  b[0] = wpack[k * 128 + col]; b[1] = wpack[(k + 1) * 128 + col];
    acc = __builtin_amdgcn_wmma_f32_16x16x4_f32(
        /*neg_a=*/false, a, /*neg_b=*/false, b,
        /*c_mod=*/(short)0, acc, /*reuse_a=*/false, /*reuse_b=*/false);
  }

  // D: lane holds (M = r + 8*half, N = mn) -> straight-line store clause
  float* __restrict__ dstp = proj + (size_t)(rowbase + 8 * half) * 128 + col;
#pragma unroll
  for (int r = 0; r < 8; ++r) dstp[(size_t)r * 128] = acc[r];
}

// ------------------------------------------------------------- edge pass 1
// alpha[e,h] = (q[dst,h,:] . (k[src,h,:] + edge_attr[e]@We[:,h,:])) / sqrt(C)
// + segment max via order-preserving uint atomicMax (sentinel 0 = empty).
__device__ __forceinline__ unsigned int gt_ordenc(float f) {
  unsigned int u = __float_as_uint(f);
  return (u & 0x80000000u) ? ~u : (u | 0x80000000u);
}
__device__ __forceinline__ float gt_orddec(unsigned int e) {
  return __uint_as_float((e & 0x80000000u) ? (e & 0x7FFFFFFFu) : ~e);
}

template <int HC, int C>
__global__ void gt_edge_alpha_kernel(const float* __restrict__ proj,
                                     const int* __restrict__ ei,
                                     const float* __restrict__ eattr,
                                     const float* __restrict__ weRows,  // 4 rows, stride 128
                                     float* __restrict__ alpha,
                                     unsigned int* __restrict__ amax, int Etot) {
  __shared__ float sWe[4 * HC];
  for (int t = threadIdx.x; t < 4 * HC; t += blockDim.x)
    sWe[t] = weRows[(t / HC) * 128 + (t % HC)];
  __syncthreads();

  const int e = blockIdx.x * blockDim.x + threadIdx.x;
  if (e >= Etot) return;
  const int s = ei[e];
  const int d = ei[Etot + e];
  const float ea0 = eattr[e * 4 + 0], ea1 = eattr[e * 4 + 1];
  const float ea2 = eattr[e * 4 + 2], ea3 = eattr[e * 4 + 3];
  const float* __restrict__ ps = proj + (size_t)s * 128;  // [q|k|v|skip]
  const float* __restrict__ pd = proj + (size_t)d * 128;
  const float scale = (C == 4) ? 0.5f : 0.3535533905932738f; // 1/sqrt(C)

#pragma unroll
  for (int h = 0; h < 4; ++h) {
    float dot = 0.0f;
#pragma unroll
    for (int i = 0; i < C; ++i) {
      const int j = h * C + i;
      const float em = ea0 * sWe[j] + ea1 * sWe[HC + j] +
                       ea2 * sWe[2 * HC + j] + ea3 * sWe[3 * HC + j];
      dot += pd[j] * (ps[HC + j] + em);
    }
    const float a = dot * scale;
    alpha[(size_t)e * 4 + h] = a;
    atomicMax(&amax[(size_t)d * 4 + h], gt_ordenc(a));
  }
}

// ------------------------------------------------------------- edge pass 2
// ex = exp(alpha - amax[dst]); den[dst,h] += ex; acc[dst,:] += ex * (v[src]+e)
// (normalization by den folded into the node finalize pass)
template <int HC, int C>
__global__ void gt_edge_scatter_kernel(const float* __restrict__ proj,
                                       const int* __restrict__ ei,
                                       const float* __restrict__ eattr,
                                       const float* __restrict__ weRows,
                                       const float* __restrict__ alpha,
                                       const unsigned int* __restrict__ amax,
                                       float* __restrict__ den,
                                       float* __restrict__ acc, int Etot) {
  __shared__ float sWe[4 * HC];
  for (int t = threadIdx.x; t < 4 * HC; t += blockDim.x)
    sWe[t] = weRows[(t / HC) * 128 + (t % HC)];
  __syncthreads();

  const int e = blockIdx.x * blockDim.x + threadIdx.x;
  if (e >= Etot) return;
  const int s = ei[e];
  const int d = ei[Etot + e];
  const float ea0 = eattr[e * 4 + 0], ea1 = eattr[e * 4 + 1];
  const float ea2 = eattr[e * 4 + 2], ea3 = eattr[e * 4 + 3];
  const float* __restrict__ ps = proj + (size_t)s * 128;

#pragma unroll
  for (int h = 0; h < 4; ++h) {
    const unsigned int enc = amax[(size_t)d * 4 + h];
    const float am = (enc == 0u) ? 0.0f : gt_orddec(enc);
    const float ex = __expf(alpha[(size_t)e * 4 + h] - am);
    unsafeAtomicAdd(&den[(size_t)d * 4 + h], ex);
#pragma unroll
    for (int i = 0; i < C; ++i) {
      const int j = h * C + i;
      const float em = ea0 * sWe[j] + ea1 * sWe[HC + j] +
                       ea2 * sWe[2 * HC + j] + ea3 * sWe[3 * HC + j];
      const float vj = ps[2 * HC + j] + em;
      unsafeAtomicAdd(&acc[(size_t)d * HC + j], ex * vj);
    }
  }
}

// --------------------------------------------------------------- finalizers
// layers 0,1: concat heads, /= den, + skip(col 96..127), PReLU
__global__ void gt_finalize_concat_kernel(const float* __restrict__ acc,
                                          const float* __restrict__ den,
                                          const float* __restrict__ proj,
                                          float* __restrict__ hnext,
                                          const float* __restrict__ prelu_a, int total) {
  int idx = blockIdx.x * blockDim.x + threadIdx.x;
  if (idx >= total) return;
  const int n = idx >> 5;        // /32
  const int j = idx & 31;
  const int h = j >> 3;          // C = 8
  float o = acc[idx] / (den[(size_t)n * 4 + h] + 1e-16f) +
            proj[(size_t)n * 128 + 96 + j];
  const float a = *prelu_a;
  hnext[idx] = (o >= 0.0f) ? o : a * o;
}

// layer 2: mean over heads (C=4), + skip(col 48..51)
__global__ void gt_finalize_mean_kernel(const float* __restrict__ acc,
                                        const float* __restrict__ den,
                                        const float* __restrict__ proj,
                                        float* __restrict__ out, int total) {
  int idx = blockIdx.x * blockDim.x + threadIdx.x;
  if (idx >= total) return;
  const int n = idx >> 2;
  const int j = idx & 3;
  float s = 0.0f;
#pragma unroll
  for (int h = 0; h < 4; ++h)
    s += acc[(size_t)n * 16 + h * 4 + j] / (den[(size_t)n * 4 + h] + 1e-16f);
  out[idx] = 0.25f * s + proj[(size_t)n * 128 + 48 + j];
}

// ------------------------------------------------------------------ driver
extern "C" void kernel_launch(void* const* d_in, const int* in_sizes, int n_in,
                              void* d_out, int out_size, void* d_ws, size_t ws_size,
                              hipStream_t stream) {
  (void)in_sizes; (void)out_size; (void)ws_size;

  const float* x     = (const float*)d_in[0];
  const int*   ei    = (const int*)d_in[1];
  const float* eattr = (const float*)d_in[2];
  // params: d_in[3+9l .. 11+9l] = Wq,bq,Wk,bk,Wv,bv,We,Wskip,bskip
  const float* pr0 = (const float*)d_in[30];
  const float* pr1 = (n_in > 31) ? (const float*)d_in[31] : pr0 + 1;

  float* ws    = (float*)d_ws;
  float* proj  = ws;                                 // N*128
  float* hA    = proj + (size_t)NN * 128;            // N*32
  float* hB    = hA + (size_t)NN * 32;               // N*32
  float* alpha = hB + (size_t)NN * 32;               // E*4
  unsigned int* amax = (unsigned int*)(alpha + (size_t)EE * 4);  // N*4
  float* den   = (float*)(amax + (size_t)NN * 4);    // N*4   } contiguous with
  float* acc   = den + (size_t)NN * 4;               // N*32  } amax for one memset
  float* wpack = acc + (size_t)NN * 32;              // 37*128

  const int dins[3]  = {16, 32, 32};
  const int hcs[3]   = {32, 32, 16};
  const int skips[3] = {32, 32, 4};

  const float* hin = x;
  for (int l = 0; l < 3; ++l) {
    const float* const* P = (const float* const*)(d_in + 3 + 9 * l);
    gt_pack_kernel<<<(37 * 128 + 255) / 256, 256, 0, stream>>>(
        wpack, P[0], P[1], P[2], P[3], P[4], P[5], P[6], P[7], P[8],
        dins[l], hcs[l], skips[l]);

    if (dins[l] == 16)
      gt_proj_wmma_kernel<16><<<NN / 16, 256, 0, stream>>>(hin, wpack, proj);
    else
      gt_proj_wmma_kernel<32><<<NN / 16, 256, 0, stream>>>(hin, wpack, proj);

    gt_zero_kernel<<<(NN * 40 + 255) / 256, 256, 0, stream>>>((float*)amax, NN * 40);

    const float* weRows = wpack + 33 * 128;
    if (l < 2) {
      gt_edge_alpha_kernel<32, 8><<<(EE + 255) / 256, 256, 0, stream>>>(
          proj, ei, eattr, weRows, alpha, amax, EE);
      gt_edge_scatter_kernel<32, 8><<<(EE + 255) / 256, 256, 0, stream>>>(
          proj, ei, eattr, weRows, alpha, amax, den, acc, EE);
      float* hout = (l == 0) ? hA : hB;
      gt_finalize_concat_kernel<<<(NN * 32 + 255) / 256, 256, 0, stream>>>(
          acc, den, proj, hout, (l == 0) ? pr0 : pr1, NN * 32);
      hin = hout;
    } else {
      gt_edge_alpha_kernel<16, 4><<<(EE + 255) / 256, 256, 0, stream>>>(
          proj, ei, eattr, weRows, alpha, amax, EE);
      gt_edge_scatter_kernel<16, 4><<<(EE + 255) / 256, 256, 0, stream>>>(
          proj, ei, eattr, weRows, alpha, amax, den, acc, EE);
      gt_finalize_mean_kernel<<<(NN * 4 + 255) / 256, 256, 0, stream>>>(
          acc, den, proj, (float*)d_out, NN * 4);
    }
  }
}